// MicroscopyIRadon_35751307771956
// MI455X (gfx1250) — compile-verified
//
#include <hip/hip_runtime.h>
#include <stdint.h>

#define W      512
#define NANG   180
#define CH     20                       // angles per LDS chunk
#define NCHUNK (NANG / CH)              // 9
#define TPB    256                      // 8 waves (wave32)
#define LPC    ((CH * W * 4) / (16 * TPB))  // b128 async loads per thread per chunk = 10

static_assert(LPC == 10, "load count mismatch");
static_assert(NANG % CH == 0, "chunking mismatch");

__global__ __launch_bounds__(TPB)
void MicroscopyIRadon_bp_kernel(const float* __restrict__ sino,
                                const float* __restrict__ angles,
                                float* __restrict__ out) {
    __shared__ __align__(16) float s_sino[2][CH * W];   // 80 KB double buffer
    __shared__ float s_ca[NANG];                        // cos(th) * 255.5
    __shared__ float s_sa[NANG];                        // sin(th) * 255.5

    const int tid = (int)threadIdx.x;

    // Trig tables (deg -> rad), pre-scaled by 0.5*(W-1); built once per block.
    for (int n = tid; n < NANG; n += TPB) {
        float th = angles[n] * 0.017453292519943295f;
        s_ca[n] = cosf(th) * 255.5f;
        s_sa[n] = sinf(th) * 255.5f;
    }

    // Pixel assignment: 32x8 tile per block, lane-consecutive in x.
    const int x = (int)blockIdx.x * 32 + (tid & 31);
    const int y = (int)blockIdx.y * 8 + (tid >> 5);
    const float xs = (float)x * (2.0f / (float)(W - 1)) - 1.0f;
    const float ys = (float)y * (2.0f / (float)(W - 1)) - 1.0f;

    // Async global->LDS stage of one angle chunk (ASYNCcnt path).
    auto issue_chunk = [&](int c, int buf) {
        const uint64_t gbase = (uint64_t)(uintptr_t)(sino + (size_t)c * CH * W);
        const uint32_t lbase = (uint32_t)(uintptr_t)(&s_sino[buf][0]); // low 32 bits = LDS byte addr
#pragma unroll
        for (int t = 0; t < LPC; ++t) {
            const uint32_t boff = (uint32_t)(tid + t * TPB) * 16u;
            const uint32_t ldsa = lbase + boff;
            asm volatile("global_load_async_to_lds_b128 %0, %1, %2"
                         :
                         : "v"(ldsa), "v"(boff), "s"(gbase)
                         : "memory");
        }
    };

    float acc = 0.0f;

    issue_chunk(0, 0);

    for (int c = 0; c < NCHUNK; ++c) {
        const int buf = c & 1;
        if (c + 1 < NCHUNK) {
            issue_chunk(c + 1, buf ^ 1);
            // 10 loads outstanding for chunk c+1; async loads complete in order,
            // so asynccnt<=10 means chunk c's data has landed in LDS.
            asm volatile("s_wait_asynccnt 10" ::: "memory");
        } else {
            asm volatile("s_wait_asynccnt 0" ::: "memory");
        }
        __syncthreads();   // all waves' chunk-c loads visible

        const float* __restrict__ srow = &s_sino[buf][0];
#pragma unroll 5
        for (int a = 0; a < CH; ++a) {
            const int n = c * CH + a;
            const float ca = s_ca[n];
            const float sa = s_sa[n];
            // ix = (gx+1)*0.5*(W-1) = ca*xs + sa*ys + 255.5   (ca,sa pre-scaled)
            const float ix = fmaf(ca, xs, fmaf(sa, ys, 255.5f));
            const float iy = fmaf(ca, ys, fmaf(-sa, xs, 255.5f));

            const float x0f = floorf(ix);
            const float wx1 = ix - x0f;
            const float wx0 = 1.0f - wx1;
            const int   x0i = min(max((int)x0f, 0), W - 1);
            const int   x1i = min(max((int)x0f + 1, 0), W - 1);

            const float y0f = floorf(iy);
            const float wy1 = iy - y0f;
            const float wy0 = 1.0f - wy1;
            const float yw  = ((y0f >= 0.0f  && y0f <= 511.0f) ? wy0 : 0.0f)
                            + ((y0f >= -1.0f && y0f <= 510.0f) ? wy1 : 0.0f);

            // Fold x-masks and yw into the two tap weights.
            const float wA = ((x0f >= 0.0f  && x0f <= 511.0f) ? wx0 : 0.0f) * yw;
            const float wB = ((x0f >= -1.0f && x0f <= 510.0f) ? wx1 : 0.0f) * yw;

            const float g0 = srow[a * W + x0i];   // ds_load_b32, ~stride-|cos| across lanes
            const float g1 = srow[a * W + x1i];
            acc = fmaf(wA, g0, fmaf(wB, g1, acc));
        }
        __syncthreads();   // buffer consumed; safe for the overwrite two chunks ahead
    }

    out[(size_t)y * W + x] = acc * (1.0f / (float)NANG);
}

extern "C" void kernel_launch(void* const* d_in, const int* in_sizes, int n_in,
                              void* d_out, int out_size, void* d_ws, size_t ws_size,
                              hipStream_t stream) {
    const float* sino   = (const float*)d_in[0];   // (1, 180, 512) f32
    const float* angles = (const float*)d_in[1];   // (180,) f32 degrees
    float* out          = (float*)d_out;           // (1, 1, 512, 512) f32

    dim3 grid(W / 32, W / 8);   // 16 x 64 blocks, 32x8 pixel tiles
    MicroscopyIRadon_bp_kernel<<<grid, TPB, 0, stream>>>(sino, angles, out);
    (void)in_sizes; (void)n_in; (void)out_size; (void)d_ws; (void)ws_size;
}